// LDetection_12103217840297
// MI455X (gfx1250) — compile-verified
//
#include <hip/hip_runtime.h>
#include <hip/hip_bf16.h>
#include <math.h>

// ATSS anchor assignment, two passes, no NxM materialization.
// Pass A uses V_WMMA_F32_16X16X4_F32: squared center distance is an exact
// rank-4 product  d^2 = [ax,ay,|a|^2,1] . [-2gx,-2gy,1,|g|^2]^T.
// Pass A is segmented over the anchor dimension (S segments, chosen at launch
// from ws_size) for wave-count parallelism; a small merge kernel reduces the
// S partial top-9 lists per GT and computes thresh = mean + std(ddof=1).

#define TOPK 9
#define WAVES_PER_BLOCK 8  // 256 threads = 8 wave32

typedef __attribute__((ext_vector_type(2))) float v2f;
typedef __attribute__((ext_vector_type(8))) float v8f;

// Sorted-descending 9-element top-k (smallest-9 kept; kd[0] = worst kept).
__device__ __forceinline__ void topk_insert(float (&kd)[TOPK], int (&ki)[TOPK],
                                            float x, int idx) {
  if (x < kd[0]) {
    kd[0] = x; ki[0] = idx;
#pragma unroll
    for (int j = 0; j < TOPK - 1; ++j) {
      if (kd[j] < kd[j + 1]) {
        float td = kd[j]; kd[j] = kd[j + 1]; kd[j + 1] = td;
        int   ti = ki[j]; ki[j] = ki[j + 1]; ki[j + 1] = ti;
      }
    }
  }
}

__device__ __forceinline__ float box_iou(float4 a, float4 g) {
  float lx = fmaxf(a.x, g.x), ly = fmaxf(a.y, g.y);
  float rx = fminf(a.z, g.z), ry = fminf(a.w, g.w);
  float w = fmaxf(rx - lx, 0.0f), h = fmaxf(ry - ly, 0.0f);
  float inter  = w * h;
  float area_a = (a.z - a.x) * (a.w - a.y);
  float area_g = (g.z - g.x) * (g.w - g.y);
  float uni    = area_a + area_g - inter;
  return inter / fmaxf(uni, 1e-12f);
}

// mean + std(ddof=1) of the IoUs of the 9 selected anchors vs GT g.
__device__ __forceinline__ float thresh_from_topk(
    const float* __restrict__ anchors, const float* __restrict__ gt,
    const int (&fi)[TOPK], int g, int N) {
  float4 gb = ((const float4*)gt)[g];
  float vals[TOPK];
  float sum = 0.0f;
#pragma unroll
  for (int j = 0; j < TOPK; ++j) {
    int a = fi[j];
    float4 ab = ((const float4*)anchors)[(a >= 0 && a < N) ? a : 0];
    vals[j] = box_iou(ab, gb);
    sum += vals[j];
  }
  float mean = sum * (1.0f / TOPK);
  float var = 0.0f;
#pragma unroll
  for (int j = 0; j < TOPK; ++j) { float t = vals[j] - mean; var += t * t; }
  var *= 1.0f / (TOPK - 1);
  return mean + sqrtf(var);
}

// ---------------------------------------------------------------------------
// Pass A scan: block (gtile, seg) covers 16 GTs over anchor chunks
// c = seg*8 + wave (mod S*8). One f32 WMMA per 16x16 d^2 tile; a min-tree over
// the 8 outputs gates the insertion path (common case: whole chunk rejected
// with one compare). The 16 half-lane top-9 lists are merged in LDS by 16
// threads; result goes to the partial-list workspace (direct==0) or straight
// to thresh (direct==1, S==1).
// ---------------------------------------------------------------------------
__global__ __launch_bounds__(256) void atss_scan_kernel(
    const float* __restrict__ anchors, const float* __restrict__ gt,
    float* __restrict__ pd, int* __restrict__ pi, float* __restrict__ thresh,
    int N, int M, int S, int direct) {
  const int lane = threadIdx.x & 31;
  const int wave = threadIdx.x >> 5;         // 0..7
  const int half = (lane >> 4) & 1;          // lane half selects K=0,1 vs K=2,3
  const int gl   = lane & 15;                // GT within tile
  const int g    = blockIdx.x * 16 + gl;     // global GT id
  const int seg  = blockIdx.y;               // anchor segment

  __shared__ float sd[16][16][TOPK];         // [gt_local][wave*2+half][k]
  __shared__ int   si[16][16][TOPK];

  // B-matrix operand (loop invariant): column gl = GT feature vector.
  float gx, gy;
  {
    int gg = (g < M) ? g : (M - 1);
    float4 gb = ((const float4*)gt)[gg];
    gx = 0.5f * (gb.x + gb.z) - 512.0f;      // center-shift for f32 accuracy
    gy = 0.5f * (gb.y + gb.w) - 512.0f;
  }
  v2f bmat;
  if (half == 0) { bmat[0] = -2.0f * gx; bmat[1] = -2.0f * gy; }      // K=0,1
  else           { bmat[0] = 1.0f;       bmat[1] = gx * gx + gy * gy; } // K=2,3

  float kd[TOPK]; int ki[TOPK];
#pragma unroll
  for (int j = 0; j < TOPK; ++j) { kd[j] = 3.0e38f; ki[j] = 0; }

  const int nch = (N + 15) >> 4;             // 16-anchor chunks
  for (int c = seg * WAVES_PER_BLOCK + wave; c < nch; c += S * WAVES_PER_BLOCK) {
    const int abase = c * 16;
    const int aidx  = abase + gl;
    float ax, ay;
    if (aidx < N) {
      float4 ab = ((const float4*)anchors)[aidx];
      ax = 0.5f * (ab.x + ab.z) - 512.0f;
      ay = 0.5f * (ab.y + ab.w) - 512.0f;
    } else {                                  // pad row -> astronomically far
      ax = 1.0e8f; ay = 1.0e8f;
    }
    v2f amat;
    if (half == 0) { amat[0] = ax; amat[1] = ay; }                  // K=0,1
    else           { amat[0] = ax * ax + ay * ay; amat[1] = 1.0f; } // K=2,3

    v8f cz = {0.f, 0.f, 0.f, 0.f, 0.f, 0.f, 0.f, 0.f};
    // D[m][n] = d^2(anchor abase+m, gt tile col n)
    v8f d2 = __builtin_amdgcn_wmma_f32_16x16x4_f32(
        false, amat, false, bmat, (short)0, cz, false, false);

    // Min-tree gate: one compare rejects the whole chunk in the common case.
    float mn = fminf(fminf(fminf(d2[0], d2[1]), fminf(d2[2], d2[3])),
                     fminf(fminf(d2[4], d2[5]), fminf(d2[6], d2[7])));
    if (mn < kd[0]) {
#pragma unroll
      for (int v = 0; v < 8; ++v) {
        int row = half ? (8 + v) : v;        // D layout: lanes>=16 hold M=8..15
        topk_insert(kd, ki, d2[v], abase + row);
      }
    }
  }

  const int slot = wave * 2 + half;
#pragma unroll
  for (int j = 0; j < TOPK; ++j) { sd[gl][slot][j] = kd[j]; si[gl][slot][j] = ki[j]; }
  __syncthreads();

  if (threadIdx.x < 16) {                    // thread t owns GT local t (== gl)
    float fd[TOPK]; int fi[TOPK];
#pragma unroll
    for (int j = 0; j < TOPK; ++j) { fd[j] = 3.0e38f; fi[j] = 0; }
    for (int s = 0; s < 16; ++s)
      for (int j = 0; j < TOPK; ++j)
        topk_insert(fd, fi, sd[gl][s][j], si[gl][s][j]);

    if (g < M) {
      if (direct) {
        thresh[g] = thresh_from_topk(anchors, gt, fi, g, N);
      } else {
        size_t base = ((size_t)g * S + seg) * TOPK;
#pragma unroll
        for (int j = 0; j < TOPK; ++j) { pd[base + j] = fd[j]; pi[base + j] = fi[j]; }
      }
    }
  }
}

// ---------------------------------------------------------------------------
// Pass A merge: one thread per GT merges S partial top-9 lists -> final top-9,
// then computes thresh[g].
// ---------------------------------------------------------------------------
__global__ __launch_bounds__(256) void atss_merge_kernel(
    const float* __restrict__ anchors, const float* __restrict__ gt,
    const float* __restrict__ pd, const int* __restrict__ pi,
    float* __restrict__ thresh, int N, int M, int S) {
  const int g = blockIdx.x * blockDim.x + threadIdx.x;
  if (g >= M) return;
  float fd[TOPK]; int fi[TOPK];
#pragma unroll
  for (int j = 0; j < TOPK; ++j) { fd[j] = 3.0e38f; fi[j] = 0; }
  const size_t base = (size_t)g * S * TOPK;
  for (int t = 0; t < S * TOPK; ++t)
    topk_insert(fd, fi, pd[base + t], pi[base + t]);
  thresh[g] = thresh_from_topk(anchors, gt, fi, g, N);
}

// ---------------------------------------------------------------------------
// Pass B: one thread per anchor; GT boxes packed as float4 + (area,thr) float2
// in LDS -> ds_load_b128 + ds_load_b64 per GT iteration. Division-free inner
// loop: iou >= thr  <=>  inter >= thr * max(union, 1e-12). Ascending m loop ->
// last match == max gt index (reference semantics); one divide at the end.
// ---------------------------------------------------------------------------
__global__ __launch_bounds__(256) void atss_assign_kernel(
    const float* __restrict__ anchors, const float* __restrict__ gt,
    const float* __restrict__ thresh, float* __restrict__ out, int N, int M) {
  extern __shared__ float smraw[];
  float4* sbox = (float4*)smraw;             // M boxes
  float2* sat  = (float2*)(smraw + 4 * M);   // M x (area, thresh)

  for (int m = threadIdx.x; m < M; m += blockDim.x) {
    float4 g = ((const float4*)gt)[m];
    sbox[m] = g;
    sat[m]  = make_float2((g.z - g.x) * (g.w - g.y), thresh[m]);
  }
  __syncthreads();

  const int n = blockIdx.x * blockDim.x + threadIdx.x;
  if (n >= N) return;

  float4 a = ((const float4*)anchors)[n];
  float cx = 0.5f * (a.x + a.z), cy = 0.5f * (a.y + a.w);
  float area_a = (a.z - a.x) * (a.w - a.y);

  int matched = -1;
  float bi = 0.0f, bu = 1.0f;                // inter/union at the match
  for (int m = 0; m < M; ++m) {
    float4 gb = sbox[m];                     // ds_load_b128 (broadcast)
    float2 at = sat[m];                      // ds_load_b64  (broadcast)
    float lx = fmaxf(a.x, gb.x), ly = fmaxf(a.y, gb.y);
    float rx = fminf(a.z, gb.z), ry = fminf(a.w, gb.w);
    float w = fmaxf(rx - lx, 0.0f), h = fmaxf(ry - ly, 0.0f);
    float inter = w * h;
    float maxu  = fmaxf(area_a + at.x - inter, 1e-12f);
    bool inside = (cx >= gb.x) & (cx <= gb.z) & (cy >= gb.y) & (cy <= gb.w);
    if (inside && inter >= at.y * maxu) { matched = m; bi = inter; bu = maxu; }
  }
  out[n]     = (float)matched;               // matched_idxs (<= 799, exact f32)
  out[N + n] = (matched >= 0) ? (bi / bu) : 0.0f;  // max_ious
}

extern "C" void kernel_launch(void* const* d_in, const int* in_sizes, int n_in,
                              void* d_out, int out_size, void* d_ws, size_t ws_size,
                              hipStream_t stream) {
  const float* anchors = (const float*)d_in[0];
  const float* gt      = (const float*)d_in[1];
  const int N = in_sizes[0] / 4;
  const int M = in_sizes[1] / 4;

  float* thresh = (float*)d_ws;
  float* out    = (float*)d_out;

  const size_t threshBytes = (size_t)M * sizeof(float);
  auto need = [&](int s) {
    return threshBytes + (size_t)M * s * TOPK * (sizeof(float) + sizeof(int));
  };

  // Pick the largest segment count the workspace allows (deterministic:
  // ws_size is fixed across calls). Fall back to fully in-kernel thresh
  // computation if even S=1 partial lists don't fit.
  int S = 16;
  while (S > 1 && need(S) > ws_size) S >>= 1;
  const int direct = (need(1) > ws_size) ? 1 : 0;
  if (direct) S = 1;

  float* pd = (float*)((char*)d_ws + threshBytes);
  int*   pi = (int*)((char*)pd + (size_t)M * S * TOPK * sizeof(float));

  dim3 blockA(256), gridA((M + 15) / 16, S);
  atss_scan_kernel<<<gridA, blockA, 0, stream>>>(anchors, gt, pd, pi, thresh,
                                                 N, M, S, direct);
  if (!direct) {
    dim3 blockM(256), gridM((M + 255) / 256);
    atss_merge_kernel<<<gridM, blockM, 0, stream>>>(anchors, gt, pd, pi,
                                                    thresh, N, M, S);
  }

  dim3 blockB(256), gridB((N + 255) / 256);
  size_t smemB = (size_t)M * 6 * sizeof(float);  // float4 + float2 per GT
  atss_assign_kernel<<<gridB, blockB, smemB, stream>>>(anchors, gt, thresh,
                                                       out, N, M);
}